// KoopmanDynamics_30442728194405
// MI455X (gfx1250) — compile-verified
//
#include <hip/hip_runtime.h>
#include <hip/hip_bf16.h>
#include <math.h>

// Problem constants (match reference)
#define BSZ 1024
#define DD  64
#define RR  8
#define GG  32
#define HH  256
#define KST 16
#define LD  68   // padded LDS row stride for 64x64 matrices (bank-conflict-free)

typedef float v2f __attribute__((ext_vector_type(2)));
typedef float v8f __attribute__((ext_vector_type(8)));

__device__ __forceinline__ float silu_f(float x) {
    return x * (1.0f / (1.0f + __expf(-x)));
}

// -------------------------------------------------------------------------
// 64x64x64 GEMM in LDS using V_WMMA_F32_16X16X4_F32, with the accumulator
// pre-initialized to diagC*I (fuses "c*I + M*X" Horner steps into the WMMA
// accumulate — no separate elementwise pass or extra barrier needed).
// 16 waves (512 threads): wave w computes output tile (w>>2, w&3).
// A,B,C are LDS matrices with row stride LD.
// A-frag (16x4 f32): lane L -> m=L%16, kb=2*(L/16); v0=A[m][kb], v1=A[m][kb+1]
// B-frag (4x16 f32): lane L -> n=L%16, kb=2*(L/16); v0=B[kb][n], v1=B[kb+1][n]
// C/D  (16x16 f32):  lane L -> n=L%16, half=L/16; vgpr v holds row v+8*half
// -------------------------------------------------------------------------
__device__ __forceinline__ void gemm64_diag(const float* __restrict__ A,
                                            const float* __restrict__ B,
                                            float* __restrict__ C,
                                            float diagC) {
    const int tid  = threadIdx.x;
    const int wave = tid >> 5;
    const int lane = tid & 31;
    const int ti   = (wave >> 2) << 4;   // tile row base
    const int tj   = (wave & 3) << 4;    // tile col base
    const int mn   = lane & 15;
    const int half = lane >> 4;
    const int kb   = half << 1;

    const float* Ap = A + (ti + mn) * LD + kb;  // A row, k-offset by lane half
    const float* Bp = B + kb * LD + tj + mn;    // B col, k-offset by lane half

    v8f acc = {};
    if (ti == tj) {   // uniform per wave: diagonal tile gets diagC on its diag
#pragma unroll
        for (int v = 0; v < 8; ++v)
            acc[v] = ((v + 8 * half) == mn) ? diagC : 0.0f;
    }
#pragma unroll
    for (int k = 0; k < 64; k += 4) {
        v2f av, bv;
        av.x = Ap[k];
        av.y = Ap[k + 1];
        bv.x = Bp[k * LD];
        bv.y = Bp[k * LD + LD];
        acc = __builtin_amdgcn_wmma_f32_16x16x4_f32(
            false, av, false, bv, (short)0, acc, false, false);
    }
#pragma unroll
    for (int v = 0; v < 8; ++v) {
        C[(ti + v + 8 * half) * LD + tj + mn] = acc[v];
    }
}

// -------------------------------------------------------------------------
// Phase 1: build K[b] = 0.5*(A - A^T) - L L^T - gamma*I   for each batch b.
// 1024 blocks x 256 threads.
// -------------------------------------------------------------------------
__global__ __launch_bounds__(256) void koop_prepare_K(
    const float* __restrict__ g,
    const float* __restrict__ sw1, const float* __restrict__ sb1,
    const float* __restrict__ sw2, const float* __restrict__ sb2,
    const float* __restrict__ dw1, const float* __restrict__ db1,
    const float* __restrict__ dw2, const float* __restrict__ db2,
    const float* __restrict__ gamma_p,
    float* __restrict__ Kg) {
    __shared__ float gs[GG];
    __shared__ float h1s[HH];
    __shared__ float h1d[HH];
    __shared__ float As[DD * 65];   // padded to avoid transpose bank conflicts
    __shared__ float Ls[DD * RR];

    const int tid = threadIdx.x;
    const int b   = blockIdx.x;

    if (tid < GG) gs[tid] = g[b * GG + tid];
    __syncthreads();

    {   // hidden layers of skew and dissipation MLPs
        float a = sb1[tid], d = db1[tid];
#pragma unroll
        for (int i = 0; i < GG; ++i) {
            float gv = gs[i];
            a = fmaf(gv, sw1[i * HH + tid], a);
            d = fmaf(gv, dw1[i * HH + tid], d);
        }
        h1s[tid] = silu_f(a);
        h1d[tid] = silu_f(d);
    }
    __syncthreads();

    // A = h1s @ sw2 + sb2  (4096 outputs, 16 per thread, coalesced in o)
#pragma unroll 1
    for (int c = 0; c < 16; ++c) {
        int o = tid + c * 256;
        float acc = sb2[o];
        for (int h = 0; h < HH; ++h) acc = fmaf(h1s[h], sw2[h * 4096 + o], acc);
        As[(o >> 6) * 65 + (o & 63)] = acc;
    }
    // L = h1d @ dw2 + db2  (512 outputs, 2 per thread)
#pragma unroll 1
    for (int c = 0; c < 2; ++c) {
        int o = tid + c * 256;
        float acc = db2[o];
        for (int h = 0; h < HH; ++h) acc = fmaf(h1d[h], dw2[h * 512 + o], acc);
        Ls[o] = acc;
    }
    __syncthreads();

    float gp = gamma_p[0];
    float gamma = ((gp > 20.0f) ? gp : log1pf(__expf(gp))) + 1e-4f;

#pragma unroll 1
    for (int c = 0; c < 16; ++c) {
        int idx = tid + c * 256;
        int i = idx >> 6, j = idx & 63;
        float v = 0.5f * (As[i * 65 + j] - As[j * 65 + i]);
#pragma unroll
        for (int r = 0; r < RR; ++r) v = fmaf(-Ls[i * RR + r], Ls[j * RR + r], v);
        if (i == j) v -= gamma;
        Kg[b * (DD * DD) + idx] = v;
    }
}

// -------------------------------------------------------------------------
// Phase 2: dt_phys + bias MLP -> bias_vec[b][k][0:64], dt_phys[b][k].
// 16384 blocks (one per (b,k)) x 256 threads.
// -------------------------------------------------------------------------
__global__ __launch_bounds__(256) void koop_bias(
    const float* __restrict__ dt_norm, const float* __restrict__ g,
    const float* __restrict__ bw1, const float* __restrict__ bb1,
    const float* __restrict__ bw2, const float* __restrict__ bb2,
    const float* __restrict__ bw3, const float* __restrict__ bb3,
    float* __restrict__ bias_vec, float* __restrict__ dt_phys_g) {
    __shared__ float in[2 + GG];
    __shared__ float h1[HH];
    __shared__ float h2[HH];
    __shared__ float sdtp;

    const int tid = threadIdx.x;
    const int bk  = blockIdx.x;           // b*16 + k
    const int b   = bk >> 4;
    const int k   = bk & 15;

    if (tid == 0) {
        float dt = dt_norm[b * KST + k];
        bool iszero = (dt <= 0.0f);
        float dtn = fminf(fmaxf(dt, 0.0f), 1.0f);
        float logdt = -6.0f + 8.0f * dtn;
        float dtp = __expf(logdt * 2.302585092994046f);   // 10^logdt
        dtp = fminf(fmaxf(dtp, 1e-6f), 100.0f);
        if (iszero) dtp = 0.0f;
        float feat = iszero ? 0.0f : log10f(fmaxf(dtp, 1e-30f));
        in[0] = dt;
        in[1] = feat;
        sdtp = dtp;
        dt_phys_g[bk] = dtp;
    }
    if (tid < GG) in[2 + tid] = g[b * GG + tid];
    __syncthreads();

    {
        float a = bb1[tid];
#pragma unroll
        for (int i = 0; i < 2 + GG; ++i) a = fmaf(in[i], bw1[i * HH + tid], a);
        h1[tid] = silu_f(a);
    }
    __syncthreads();
    {
        float a = bb2[tid];
        for (int i = 0; i < HH; ++i) a = fmaf(h1[i], bw2[i * HH + tid], a);
        h2[tid] = silu_f(a);
    }
    __syncthreads();
    if (tid < DD) {
        float a = bb3[tid];
        for (int i = 0; i < HH; ++i) a = fmaf(h2[i], bw3[i * DD + tid], a);
        bias_vec[bk * DD + tid] = tanhf(a) * sdtp;
    }
}

// -------------------------------------------------------------------------
// Phase 3: per-batch sequential scan with fused expm/phi1 via WMMA GEMMs.
//   K resident in LDS for all 16 steps; ||K|| computed once (||K*dt|| = dt*||K||).
//   M = K*dt / 2^s
//   Horner: X = pc6*I + pc7*M (elementwise), then X <- c_j*I + M*X, j=5..0
//           (diagonal add fused into WMMA accumulator, Pb<->Tb ping-pong)
//   E = I + M*P (1 GEMM, fused diag)
//   s doublings: P <- (E*P + P)/2 ; E <- E*E
//   z <- E*z + P*bias_vec
// 1024 blocks x 512 threads (16 waves).
// -------------------------------------------------------------------------
__global__ __launch_bounds__(512) void koop_expm_scan(
    const float* __restrict__ z0, const float* __restrict__ Kg,
    const float* __restrict__ bias_vec, const float* __restrict__ dt_phys_g,
    float* __restrict__ out) {
    __shared__ float Kl[DD * DD];     // K resident across all steps
    __shared__ float Ms[DD * LD];
    __shared__ float Eb[DD * LD];
    __shared__ float Pb[DD * LD];
    __shared__ float Tb[DD * LD];
    __shared__ float zv[DD];
    __shared__ float znew[DD];
    __shared__ float bvv[DD];
    __shared__ float rowsum[DD];
    __shared__ float smK;
    __shared__ float sScale;
    __shared__ int   sS;

    const int tid = threadIdx.x;
    const int b   = blockIdx.x;

    // phi1 Taylor coefficients: c[j] = 1/(j+1)!
    const float pc0 = 1.0f,            pc1 = 0.5f,           pc2 = 1.0f / 6.0f;
    const float pc3 = 1.0f / 24.0f,    pc4 = 1.0f / 120.0f,  pc5 = 1.0f / 720.0f;
    const float pc6 = 1.0f / 5040.0f,  pc7 = 1.0f / 40320.0f;

    // load K once; init z
#pragma unroll
    for (int c = 0; c < 8; ++c) {
        int idx = tid + c * 512;
        Kl[idx] = Kg[b * (DD * DD) + idx];
    }
    if (tid < DD) zv[tid] = z0[b * DD + tid];
    __syncthreads();

    // inf-norm of K, once (per-step norm is just dt * this)
    if (tid < DD) {
        float s = 0.0f;
        for (int j = 0; j < DD; ++j) s += fabsf(Kl[tid * DD + j]);
        rowsum[tid] = s;
    }
    __syncthreads();
    if (tid == 0) {
        float m = 0.0f;
        for (int i = 0; i < DD; ++i) m = fmaxf(m, rowsum[i]);
        smK = m;
    }
    __syncthreads();
    const float mK = smK;

#pragma unroll 1
    for (int k = 0; k < KST; ++k) {
        const float dtp = dt_phys_g[b * KST + k];

        // squaring count s so that ||K*dt / 2^s|| <= 0.5 ; combined scale
        if (tid == 0) {
            float m = mK * dtp;
            int s = 0;
            if (m > 0.5f) {
                s = (int)ceilf(log2f(m * 2.0f));
                if (s < 0) s = 0;
                if (s > 30) s = 30;
            }
            sS = s;
            sScale = dtp * exp2f((float)(-s));
        }
        __syncthreads();
        const int   sq  = sS;
        const float scl = sScale;

        // Ms = K * scl   and   Horner seed X = pc6*I + pc7*Ms   (one pass)
#pragma unroll
        for (int c = 0; c < 8; ++c) {
            int idx = tid + c * 512;
            int i = idx >> 6, j = idx & 63;
            float m = Kl[idx] * scl;
            Ms[i * LD + j] = m;
            Pb[i * LD + j] = pc7 * m + ((i == j) ? pc6 : 0.0f);
        }
        __syncthreads();

        // Horner: X <- c_j*I + M*X, j=5..0 (diag fused, ping-pong Pb<->Tb)
        gemm64_diag(Ms, Pb, Tb, pc5); __syncthreads();
        gemm64_diag(Ms, Tb, Pb, pc4); __syncthreads();
        gemm64_diag(Ms, Pb, Tb, pc3); __syncthreads();
        gemm64_diag(Ms, Tb, Pb, pc2); __syncthreads();
        gemm64_diag(Ms, Pb, Tb, pc1); __syncthreads();
        gemm64_diag(Ms, Tb, Pb, pc0); __syncthreads();   // P = phi1(M) in Pb

        // E = I + M*P
        gemm64_diag(Ms, Pb, Eb, 1.0f); __syncthreads();

        // squaring/doubling (rare):  P <- (E*P + P)/2 ;  E <- E*E
#pragma unroll 1
        for (int it = 0; it < sq; ++it) {
            gemm64_diag(Eb, Pb, Tb, 0.0f);
            __syncthreads();
#pragma unroll
            for (int c = 0; c < 8; ++c) {
                int idx = tid + c * 512;
                int a = (idx >> 6) * LD + (idx & 63);
                Pb[a] = 0.5f * (Tb[a] + Pb[a]);
            }
            __syncthreads();
            gemm64_diag(Eb, Eb, Tb, 0.0f);
            __syncthreads();
#pragma unroll
            for (int c = 0; c < 8; ++c) {
                int idx = tid + c * 512;
                int a = (idx >> 6) * LD + (idx & 63);
                Eb[a] = Tb[a];
            }
            __syncthreads();
        }

        // z <- E*z + P*bias_vec ; emit
        if (tid < DD) bvv[tid] = bias_vec[(b * KST + k) * DD + tid];
        __syncthreads();
        if (tid < DD) {
            float acc = 0.0f;
            for (int j = 0; j < DD; ++j)
                acc = fmaf(Eb[tid * LD + j], zv[j],
                           fmaf(Pb[tid * LD + j], bvv[j], acc));
            znew[tid] = acc;
            out[(b * KST + k) * DD + tid] = acc;
        }
        __syncthreads();
        if (tid < DD) zv[tid] = znew[tid];
        __syncthreads();
    }
}

// -------------------------------------------------------------------------
extern "C" void kernel_launch(void* const* d_in, const int* in_sizes, int n_in,
                              void* d_out, int out_size, void* d_ws, size_t ws_size,
                              hipStream_t stream) {
    const float* z        = (const float*)d_in[0];
    const float* dt_norm  = (const float*)d_in[1];
    const float* g        = (const float*)d_in[2];
    const float* skew_w1  = (const float*)d_in[3];
    const float* skew_b1  = (const float*)d_in[4];
    const float* skew_w2  = (const float*)d_in[5];
    const float* skew_b2  = (const float*)d_in[6];
    const float* dis_w1   = (const float*)d_in[7];
    const float* dis_b1   = (const float*)d_in[8];
    const float* dis_w2   = (const float*)d_in[9];
    const float* dis_b2   = (const float*)d_in[10];
    const float* bias_w1  = (const float*)d_in[11];
    const float* bias_b1  = (const float*)d_in[12];
    const float* bias_w2  = (const float*)d_in[13];
    const float* bias_b2  = (const float*)d_in[14];
    const float* bias_w3  = (const float*)d_in[15];
    const float* bias_b3  = (const float*)d_in[16];
    const float* gamma_p  = (const float*)d_in[17];
    float* out = (float*)d_out;

    // workspace layout
    float* Kg       = (float*)d_ws;                      // 1024*4096 floats
    float* bias_vec = Kg + (size_t)BSZ * DD * DD;        // 16384*64 floats
    float* dt_phys  = bias_vec + (size_t)BSZ * KST * DD; // 16384 floats

    koop_prepare_K<<<dim3(BSZ), dim3(256), 0, stream>>>(
        g, skew_w1, skew_b1, skew_w2, skew_b2,
        dis_w1, dis_b1, dis_w2, dis_b2, gamma_p, Kg);

    koop_bias<<<dim3(BSZ * KST), dim3(256), 0, stream>>>(
        dt_norm, g, bias_w1, bias_b1, bias_w2, bias_b2,
        bias_w3, bias_b3, bias_vec, dt_phys);

    koop_expm_scan<<<dim3(BSZ), dim3(512), 0, stream>>>(
        z, Kg, bias_vec, dt_phys, out);
}